// EnsembleSpace_83133386981963
// MI455X (gfx1250) — compile-verified
//
#include <hip/hip_runtime.h>

#define TOPK 8
#define NE 64           // experts (K dim)
#define NB 128          // batch rows (M dim)
#define NTOT 1048576L   // 512*2048 (N dim)
#define EPS 1e-4f
#define TILE_COLS 64
// LDS layout: K-row pairs interleaved so each WMMA B operand {B[k][c],B[k+1][c]}
// is one aligned contiguous b64. Pair stride 160 words = 128 data + 32 pad so
// the hi half-wave (pair index +1) lands on the opposite 32-bank half.
#define PAIR_STRIDE 160

typedef __attribute__((ext_vector_type(2))) float v2f;
typedef __attribute__((ext_vector_type(8))) float v8f;

// ---------------- Kernel 1: top-k mask + softmax + eps mask ----------------
__global__ __launch_bounds__(128) void EnsembleSpace_cfg_kernel(
    const float* __restrict__ config, float* __restrict__ cfg) {
  const int row = threadIdx.x;  // 128 rows, one per thread
  float v[NE];
  #pragma unroll
  for (int e = 0; e < NE; ++e) v[e] = config[row * NE + e];

  // top-8 selection; strict '>' keeps the first (lowest) index on ties,
  // matching jax.lax.top_k ordering semantics.
  unsigned long long sel = 0ull;
  for (int t = 0; t < TOPK; ++t) {
    float best = -__builtin_huge_valf();
    int bi = 0;
    #pragma unroll
    for (int e = 0; e < NE; ++e) {
      if (!((sel >> e) & 1ull) && v[e] > best) { best = v[e]; bi = e; }
    }
    sel |= (1ull << bi);
  }

  // masked logits (zeros for non-top-k), softmax over FULL width
  float m[NE];
  float mx = -__builtin_huge_valf();
  #pragma unroll
  for (int e = 0; e < NE; ++e) {
    float x = ((sel >> e) & 1ull) ? v[e] : 0.0f;
    m[e] = x;
    mx = fmaxf(mx, x);
  }
  float Z = 0.0f;
  #pragma unroll
  for (int e = 0; e < NE; ++e) { float t = __expf(m[e] - mx); m[e] = t; Z += t; }
  #pragma unroll
  for (int e = 0; e < NE; ++e) {
    float p = m[e] / Z;
    cfg[row * NE + e] = (p < EPS) ? 0.0f : p;
  }
}

// ---------------- Kernel 2: out[128,1M] = cfg[128,64] @ kern[64,1M] --------
// Each block: one 64-column slab, ALL 128 rows -> each kernel element read
// from HBM exactly once. 8 waves x 16 rows each; K=64 as 16 steps of
// v_wmma_f32_16x16x4_f32 per 16x16 C tile, B operands via conflict-free
// aligned ds_load_b64 (LDS pre-swizzled into WMMA B layout).
__global__ __launch_bounds__(256) void EnsembleSpace_gemm_kernel(
    const float* __restrict__ cfg, const float* __restrict__ kern,
    float* __restrict__ out) {
  __shared__ float lds[(NE / 2) * PAIR_STRIDE];  // 20 KB

  const int tid = threadIdx.x;
  const long col0 = (long)blockIdx.x * TILE_COLS;

  // Stage kernel tile [64 e-rows x 64 cols] into interleaved-pair LDS layout:
  //   word(e, c) = (e>>1)*PAIR_STRIDE + 2*c + (e&1)
  #pragma unroll
  for (int r = 0; r < 4; ++r) {
    int idx = tid * 4 + r * 1024;      // 4096 floats total
    int e = idx >> 6;
    int c = idx & 63;
    float4 val = *reinterpret_cast<const float4*>(kern + (size_t)e * NTOT + col0 + c);
    float* dst = &lds[(e >> 1) * PAIR_STRIDE + 2 * c + (e & 1)];
    dst[0] = val.x;
    dst[2] = val.y;
    dst[4] = val.z;
    dst[6] = val.w;
  }

  const int lane = tid & 31;
  const int wave = tid >> 5;
  const int n  = lane & 15;   // N index within 16-wide tile / M index for A rows
  const int hi = lane >> 4;   // half-wave select
  const int b_base = wave * 16;

  // A operand (cfg rows) in the ISA 32-bit 16x4 A layout:
  //   VGPR0: K = 4s + 2*hi, VGPR1: K = 4s + 2*hi + 1, row M = lane%16.
  // Contiguous 8B loads -> land directly in even-aligned pairs.
  v2f a[16];
  const float* wrow = cfg + (size_t)(b_base + n) * NE;
  #pragma unroll
  for (int s = 0; s < 16; ++s) {
    a[s] = *reinterpret_cast<const v2f*>(wrow + 4 * s + 2 * hi);
  }

  __syncthreads();

  #pragma unroll
  for (int ct = 0; ct < 4; ++ct) {
    v8f acc = {};
    const int coff = (ct * 16 + n) * 2;
    #pragma unroll
    for (int s = 0; s < 16; ++s) {
      // B operand pair {B[4s+2hi][col], B[4s+2hi+1][col]} = one aligned b64
      v2f b = *reinterpret_cast<const v2f*>(&lds[(2 * s + hi) * PAIR_STRIDE + coff]);
      acc = __builtin_amdgcn_wmma_f32_16x16x4_f32(
          /*neg_a=*/false, a[s], /*neg_b=*/false, b,
          /*c_mod=*/(short)0, acc, /*reuse_a=*/false, /*reuse_b=*/false);
    }
    // C/D layout: VGPR v -> M = v + 8*hi, N = lane%16
    #pragma unroll
    for (int v = 0; v < 8; ++v) {
      out[(size_t)(b_base + hi * 8 + v) * NTOT + col0 + ct * 16 + n] = acc[v];
    }
  }
}

extern "C" void kernel_launch(void* const* d_in, const int* in_sizes, int n_in,
                              void* d_out, int out_size, void* d_ws, size_t ws_size,
                              hipStream_t stream) {
  const float* config = (const float*)d_in[0];  // [128, 64]
  const float* kern   = (const float*)d_in[1];  // [64, 512, 2048] -> [64, 1M]
  float* out = (float*)d_out;                   // [128, 1M]
  float* cfg = (float*)d_ws;                    // 128*64 f32 scratch (32 KB)

  EnsembleSpace_cfg_kernel<<<1, 128, 0, stream>>>(config, cfg);
  EnsembleSpace_gemm_kernel<<<(int)(NTOT / TILE_COLS), 256, 0, stream>>>(cfg, kern, out);
}